// TDVP_V2_52089363366316
// MI455X (gfx1250) — compile-verified
//
#include <hip/hip_runtime.h>
#include <math.h>

#define NPOS    64
#define BATCHN  16384
#define TPB     128
#define MSTRIDE 20        // padded row stride (floats) for the per-wave M tile
#define EPSD    1e-6f

typedef __attribute__((ext_vector_type(2))) float v2f;
typedef __attribute__((ext_vector_type(8))) float v8f;

__device__ __forceinline__ void wave_lds_fence() {
  asm volatile("" ::: "memory");
  __builtin_amdgcn_wave_barrier();
  asm volatile("" ::: "memory");
}

// Build M[n] for the 32 batch elements of this wave using two
// V_WMMA_F32_16X16X4_F32 ops:  D(16x16) = XN(16x2,zero-padded K) * An(2x16).
//   A-operand 16x4 f32: VGPR0 = K0 (lanes 0-15) / K2 (lanes 16-31), VGPR1 = K1/K3
//   B-operand 4x16 f32: VGPR0 = K0 (lanes 0-15) / K2 (lanes 16-31), VGPR1 = K1/K3
//   D: VGPR v, lanes 0-15 -> row v, lanes 16-31 -> row v+8; lane%16 = column (l*4+r)
// Result is scattered to a per-wave LDS tile, then each lane reads back its own
// 4x4 matrix (row = lane) for the scalar chain recurrences.
__device__ __forceinline__ void compute_M_wave(int n, float xn0, float xn1,
                                               const float* sA, float* wm,
                                               int lane, float* Mloc) {
  const bool lo  = (lane < 16);
  const int  col = lane & 15;

  v2f bmat;
  bmat.x = lo ? sA[n * 32 + col * 2 + 0] : 0.0f;   // K=0 row | K=2 row (zero)
  bmat.y = lo ? sA[n * 32 + col * 2 + 1] : 0.0f;   // K=1 row | K=3 row (zero)

  v2f a0;                                          // batches 0..15 of the wave
  a0.x = lo ? xn0 : 0.0f;
  a0.y = lo ? xn1 : 0.0f;

  float ox0 = __shfl_xor(xn0, 16, 32);             // pull lanes 16-31 data down
  float ox1 = __shfl_xor(xn1, 16, 32);
  v2f a1;                                          // batches 16..31 of the wave
  a1.x = lo ? ox0 : 0.0f;
  a1.y = lo ? ox1 : 0.0f;

  v8f cz = {0.0f, 0.0f, 0.0f, 0.0f, 0.0f, 0.0f, 0.0f, 0.0f};
  v8f d0 = __builtin_amdgcn_wmma_f32_16x16x4_f32(false, a0, false, bmat,
                                                 (short)0, cz, false, false);
  v8f d1 = __builtin_amdgcn_wmma_f32_16x16x4_f32(false, a1, false, bmat,
                                                 (short)0, cz, false, false);

  const int half8 = (lane >> 4) << 3;              // 0 or 8
#pragma unroll
  for (int r = 0; r < 8; ++r) {
    wm[(half8 + r)      * MSTRIDE + col] = d0[r];
    wm[(half8 + r + 16) * MSTRIDE + col] = d1[r];
  }
  wave_lds_fence();
#pragma unroll
  for (int e = 0; e < 16; ++e) Mloc[e] = wm[lane * MSTRIDE + e];
  wave_lds_fence();
}

__global__ __launch_bounds__(TPB)
void tdvp_v2_kernel(const float* __restrict__ xg,
                    const float* __restrict__ Ag,
                    const float* __restrict__ Bg,
                    const float* __restrict__ scg,
                    float* __restrict__ outg,
                    float* __restrict__ wsg) {
  // A is consumed in native layout A[n][l*4+r][i]      -> 64*32 floats (8 KB)
  // B is consumed in native layout B[n][(i*4+l)][j*2+k]-> 64*64 floats (16 KB)
  __shared__ __align__(16) float sA[NPOS * 32];
  __shared__ __align__(16) float sB[NPOS * 64];
  __shared__ __align__(16) float sM[TPB / 32][32 * MSTRIDE];   // per-wave M tiles

  const int tid = threadIdx.x;
  for (int i = tid; i < NPOS * 32; i += TPB) sA[i] = Ag[i];
  for (int i = tid; i < NPOS * 64; i += TPB) sB[i] = Bg[i];
  __syncthreads();

  const int wave = tid >> 5;
  const int lane = tid & 31;
  const int b    = blockIdx.x * TPB + tid;          // one thread per batch elem
  float* wm = sM[wave];

  const float s = scg[0];
  const float* xb = xg + (size_t)b * (NPOS * 2);
  float4* ws4 = (float4*)wsg;                       // ws[n*BATCH + b] : Aright[n]

  float Mloc[16];

  // ---------------- backward pass: Aright[n] = M[n+1] ... M[63] e0 ----------
  float u0 = 0.0f, u1 = 0.0f, u2 = 0.0f, u3 = 0.0f;
  for (int n = NPOS - 1; n >= 0; --n) {
    float x0 = xb[n * 2 + 0], x1 = xb[n * 2 + 1];
    float rn = rsqrtf(x0 * x0 + x1 * x1);
    float xn0 = x0 * rn, xn1 = x1 * rn;

    compute_M_wave(n, xn0, xn1, sA, wm, lane, Mloc);

    // store normalized Aright[n]  (n==63 stores 0, matching reference vR[0])
    float nn  = sqrtf(u0 * u0 + u1 * u1 + u2 * u2 + u3 * u3) + EPSD;
    float inv = 1.0f / nn;
    ws4[(size_t)n * BATCHN + b] = make_float4(u0 * inv, u1 * inv, u2 * inv, u3 * inv);

    if (n == NPOS - 1) {                            // init: column 0 of M[63]
      u0 = Mloc[0]; u1 = Mloc[4]; u2 = Mloc[8]; u3 = Mloc[12];
    } else {                                        // u = M[n] * u
      float t0 = Mloc[0]  * u0 + Mloc[1]  * u1 + Mloc[2]  * u2 + Mloc[3]  * u3;
      float t1 = Mloc[4]  * u0 + Mloc[5]  * u1 + Mloc[6]  * u2 + Mloc[7]  * u3;
      float t2 = Mloc[8]  * u0 + Mloc[9]  * u1 + Mloc[10] * u2 + Mloc[11] * u3;
      float t3 = Mloc[12] * u0 + Mloc[13] * u1 + Mloc[14] * u2 + Mloc[15] * u3;
      u0 = t0; u1 = t1; u2 = t2; u3 = t3;
    }
  }

  // ---------------- forward pass: vL[n] = e0^T M[0] ... M[n-1], head, output -
  float v0 = 0.0f, v1 = 0.0f, v2 = 0.0f, v3 = 0.0f;
  for (int n = 0; n < NPOS; ++n) {
    float x0 = xb[n * 2 + 0], x1 = xb[n * 2 + 1];
    float rn = rsqrtf(x0 * x0 + x1 * x1);
    float xn0 = x0 * rn, xn1 = x1 * rn;

    compute_M_wave(n, xn0, xn1, sA, wm, lane, Mloc);

    // left vector: e0 at n==0 (H0 uses B[0,0,...]), else normalized vL
    float l0, l1, l2, l3;
    if (n == 0) {
      l0 = 1.0f; l1 = 0.0f; l2 = 0.0f; l3 = 0.0f;
    } else {
      float nn  = sqrtf(v0 * v0 + v1 * v1 + v2 * v2 + v3 * v3) + EPSD;
      float inv = 1.0f / nn;
      l0 = v0 * inv; l1 = v1 * inv; l2 = v2 * inv; l3 = v3 * inv;
    }

    // right vector: e0 at n==63 (Hlast uses B[63,:,0]), else stored Aright[n]
    float4 ar;
    if (n < NPOS - 1) ar = ws4[(size_t)n * BATCHN + b];
    else              ar = make_float4(1.0f, 0.0f, 0.0f, 0.0f);

    // H[j,k] = sum_{i,l} lL[i] * B[n,i,l,j,k] * ar[l]
    const float* Bn = sB + n * 64;
    float li[4] = {l0, l1, l2, l3};
    float al[4] = {ar.x, ar.y, ar.z, ar.w};
    float H00 = 0.0f, H01 = 0.0f, H10 = 0.0f, H11 = 0.0f;
#pragma unroll
    for (int i = 0; i < 4; ++i) {
#pragma unroll
      for (int l = 0; l < 4; ++l) {
        float w = li[i] * al[l];
        const float* p = Bn + (i * 4 + l) * 4;
        H00 = fmaf(w, p[0], H00);
        H01 = fmaf(w, p[1], H01);
        H10 = fmaf(w, p[2], H10);
        H11 = fmaf(w, p[3], H11);
      }
    }

    float nrm = sqrtf(H00 * H00 + H01 * H01 + H10 * H10 + H11 * H11) + EPSD;
    float scl = s / nrm;
    H00 *= scl; H01 *= scl; H10 *= scl; H11 *= scl;
    H00 = (H00 == H00) ? H00 : 0.0f;  H01 = (H01 == H01) ? H01 : 0.0f;
    H10 = (H10 == H10) ? H10 : 0.0f;  H11 = (H11 == H11) ? H11 : 0.0f;
    H00 = fmaxf(H00, 0.0f); H01 = fmaxf(H01, 0.0f);
    H10 = fmaxf(H10, 0.0f); H11 = fmaxf(H11, 0.0f);

    float y0 = H00 * xn0 + H01 * xn1;
    float y1 = H10 * xn0 + H11 * xn1;
    outg[((size_t)b * NPOS + n) * 2 + 0] = y0;
    outg[((size_t)b * NPOS + n) * 2 + 1] = y1;

    // v = v * M[n]  (row-vector update, after use)
    if (n == 0) {
      v0 = Mloc[0]; v1 = Mloc[1]; v2 = Mloc[2]; v3 = Mloc[3];
    } else {
      float t0 = v0 * Mloc[0] + v1 * Mloc[4] + v2 * Mloc[8]  + v3 * Mloc[12];
      float t1 = v0 * Mloc[1] + v1 * Mloc[5] + v2 * Mloc[9]  + v3 * Mloc[13];
      float t2 = v0 * Mloc[2] + v1 * Mloc[6] + v2 * Mloc[10] + v3 * Mloc[14];
      float t3 = v0 * Mloc[3] + v1 * Mloc[7] + v2 * Mloc[11] + v3 * Mloc[15];
      v0 = t0; v1 = t1; v2 = t2; v3 = t3;
    }
  }
}

extern "C" void kernel_launch(void* const* d_in, const int* in_sizes, int n_in,
                              void* d_out, int out_size, void* d_ws, size_t ws_size,
                              hipStream_t stream) {
  const float* x  = (const float*)d_in[0];   // (16384, 64, 2)
  const float* A  = (const float*)d_in[1];   // (64, 4, 4, 2)
  const float* B  = (const float*)d_in[2];   // (64, 4, 4, 2, 2)
  const float* sc = (const float*)d_in[3];   // (1,)
  float* out = (float*)d_out;                // (16384, 64, 2)
  float* ws  = (float*)d_ws;                 // needs 64*16384*4 floats = 16 MB

  dim3 block(TPB);
  dim3 grid(BATCHN / TPB);
  tdvp_v2_kernel<<<grid, block, 0, stream>>>(x, A, B, sc, out, ws);
}